// STD2Vformer_59158879535643
// MI455X (gfx1250) — compile-verified
//
#include <hip/hip_runtime.h>
#include <math.h>

// ---- problem constants ----
#define Bb   32
#define Cc   3
#define Nn   883
#define Ll   12
#define Pp   12
#define Dd   128
#define Mm   10
#define Hh   512
#define Kk16 16
#define DMm  5
#define NC   (Bb*Nn)        // 28256 real columns
#define NCP  28288          // padded to 442*64 for guard-free WMMA tiles
#define NL   (Nn*Ll)        // 10596
#define CNT  (Bb*Nn*Ll)     // 339072 elements per channel

typedef float v2f __attribute__((ext_vector_type(2)));
typedef float v8f __attribute__((ext_vector_type(8)));

// =====================================================================
// WMMA fp32 GEMM:  Y(Mdim x NCP) = op( W(Mdim x Kdim) @ X(Kdim x NCP) + bias )
// mode 0: store   mode 1: relu-store   mode 2: Y += W@X + bias (residual)
// one wave computes a 32(M) x 64(N) tile, K stepped by 4.
// Mdim multiple of 32, columns padded to NCP -> NO guards anywhere.
// =====================================================================
__global__ __launch_bounds__(32)
void k_wmma_gemm(const float* __restrict__ W, const float* __restrict__ X,
                 float* __restrict__ Y, const float* __restrict__ bias,
                 int Kdim, int mode)
{
  const int lane = threadIdx.x & 31;
  const int lm   = lane & 15;            // M/N index within 16
  const int kh   = (lane >> 4) << 1;     // K offset 0 or 2 (A/B frag layout)
  const int rh   = (lane >> 4) << 3;     // C/D: M = r + 8*(lane>=16)
  const int m0   = blockIdx.y * 32;
  const int n0   = blockIdx.x * 64;

  v8f acc[2][4];
  for (int mi = 0; mi < 2; ++mi)
    for (int ni = 0; ni < 4; ++ni)
      for (int r = 0; r < 8; ++r) {
        int row = m0 + mi*16 + r + rh;
        int col = n0 + ni*16 + lm;
        float v = bias[row];
        if (mode == 2) v += Y[(size_t)row*NCP + col];
        acc[mi][ni][r] = v;
      }

  const float* w0p = W + (size_t)(m0      + lm)*Kdim + kh;
  const float* w1p = W + (size_t)(m0 + 16 + lm)*Kdim + kh;
  const float* xp0 = X + (size_t)kh*NCP + n0 + lm;

  for (int k0 = 0; k0 < Kdim; k0 += 4) {
    v2f a[2];
    a[0].x = w0p[k0];  a[0].y = w0p[k0 + 1];
    a[1].x = w1p[k0];  a[1].y = w1p[k0 + 1];
    v2f bf[4];
    const float* xp = xp0 + (size_t)k0*NCP;
    for (int ni = 0; ni < 4; ++ni) {
      bf[ni].x = xp[ni*16];
      bf[ni].y = xp[ni*16 + NCP];
    }
    for (int mi = 0; mi < 2; ++mi)
      for (int ni = 0; ni < 4; ++ni)
        acc[mi][ni] = __builtin_amdgcn_wmma_f32_16x16x4_f32(
            false, a[mi], false, bf[ni], (short)0, acc[mi][ni], false, false);
  }

  for (int mi = 0; mi < 2; ++mi)
    for (int ni = 0; ni < 4; ++ni)
      for (int r = 0; r < 8; ++r) {
        int row = m0 + mi*16 + r + rh;
        int col = n0 + ni*16 + lm;
        float v = acc[mi][ni][r];
        if (mode == 1) v = fmaxf(v, 0.f);
        Y[(size_t)row*NCP + col] = v;
      }
}

// pad reg_W (36x512) -> 64x512 zero-padded, reg_b (36) -> 64
__global__ void k_pad_reg(const float* __restrict__ regW, const float* __restrict__ regb,
                          float* __restrict__ Wp, float* __restrict__ bp)
{
  int i = blockIdx.x*blockDim.x + threadIdx.x;
  if (i < 64*Hh) {
    int row = i / Hh, k = i % Hh;
    Wp[i] = (row < Cc*Ll) ? regW[row*Hh + k] : 0.f;
  }
  if (i < 64) bp[i] = (i < Cc*Ll) ? regb[i] : 0.f;
}

// =====================================================================
// per-channel BN statistics (deterministic two-level reduction)
// which==0: src layout (b,c,n,l);  which==1: src = reg (64 x NCP), regR view
// =====================================================================
__global__ __launch_bounds__(256)
void k_chan_stats(const float* __restrict__ src, float* __restrict__ partial, int which)
{
  int c = blockIdx.x, tid = threadIdx.x;
  float s = 0.f, s2 = 0.f;
  int stride = gridDim.y * blockDim.x;
  for (int i = blockIdx.y*blockDim.x + tid; i < CNT; i += stride) {
    float v;
    int b = i / NL, r = i % NL;
    if (which == 0) {
      v = src[(size_t)(b*Cc + c)*NL + r];
    } else {
      int n = r / Ll, l = r % Ll;
      v = src[(size_t)(c*Ll + l)*NCP + b*Nn + n];
    }
    s += v; s2 += v*v;
  }
  __shared__ float sh[256], sh2[256];
  sh[tid] = s; sh2[tid] = s2; __syncthreads();
  for (int o = 128; o > 0; o >>= 1) {
    if (tid < o) { sh[tid] += sh[tid+o]; sh2[tid] += sh2[tid+o]; }
    __syncthreads();
  }
  if (tid == 0) {
    partial[(c*gridDim.y + blockIdx.y)*2 + 0] = sh[0];
    partial[(c*gridDim.y + blockIdx.y)*2 + 1] = sh2[0];
  }
}

__global__ __launch_bounds__(64)
void k_stats_fin(const float* __restrict__ partial, float* __restrict__ st, int nb)
{
  int c = blockIdx.x, tid = threadIdx.x;
  __shared__ float sh[64], sh2[64];
  float s = 0.f, s2 = 0.f;
  for (int i = tid; i < nb; i += 64) {
    s  += partial[(c*nb + i)*2 + 0];
    s2 += partial[(c*nb + i)*2 + 1];
  }
  sh[tid] = s; sh2[tid] = s2; __syncthreads();
  for (int o = 32; o > 0; o >>= 1) {
    if (tid < o) { sh[tid] += sh[tid+o]; sh2[tid] += sh2[tid+o]; }
    __syncthreads();
  }
  if (tid == 0) {
    float m = sh[0] / (float)CNT;
    float var = sh2[0] / (float)CNT - m*m;
    st[c] = m; st[3+c] = rsqrtf(var + 1e-5f);
  }
}

// normalize input -> xb (b,c,n,l) and scatter to ts36 (36 x NCP)
__global__ void k_norm_in(const float* __restrict__ input, const float* __restrict__ st,
                          float* __restrict__ xb, float* __restrict__ ts36)
{
  int i = blockIdx.x*blockDim.x + threadIdx.x;
  if (i >= Bb*Cc*NL) return;
  int b = i/(Cc*NL), rem = i%(Cc*NL), c = rem/NL, r = rem%NL, n = r/Ll, l = r%Ll;
  float v = (input[i] - st[c]) * st[3+c];
  xb[i] = v;
  ts36[(size_t)(c*Ll + l)*NCP + b*Nn + n] = v;
}

// hidden rows 128..511: node / time-in-day / day-in-week embeddings
__global__ void k_fill_hidden(float* __restrict__ hidden, const float* __restrict__ mb,
                              const float* __restrict__ tde, const float* __restrict__ dwe,
                              const int* __restrict__ idx)
{
  int i = blockIdx.x*blockDim.x + threadIdx.x;
  if (i >= 384*NC) return;
  int row = 128 + i/NC, col = i%NC, b = col/Nn, n = col%Nn;
  float v;
  if      (row < 256) v = mb[n*Dd + (row-128)];
  else if (row < 384) v = tde[idx[b]*Dd + (row-256)];
  else                v = dwe[idx[32+b]*Dd + (row-384)];
  hidden[(size_t)row*NCP + col] = v;
}

__global__ void k_tidx(const float* __restrict__ seqs, int* __restrict__ idx)
{
  int b = threadIdx.x;
  if (b >= Bb) return;
  float hour   = (seqs[(b*DMm + 3)*Ll + (Ll-1)] + 0.5f) * 23.f;
  float minute = (seqs[(b*DMm + 4)*Ll + (Ll-1)] + 0.5f) * 59.f;
  int t = (int)((hour*60.f + minute) * (1.f/5.f));      // 60/PPH = 5
  t = min(max(t, 0), 287);
  int d = (int)((seqs[(b*DMm + 2)*Ll + (Ll-1)] + 0.5f) * 6.f);
  d = min(max(d, 0), 6);
  idx[b] = t; idx[32+b] = d;
}

// Date2Vec phases phi[b][t], t in 0..23
__global__ void k_phi(const float* __restrict__ seqs, const float* __restrict__ tgts,
                      const float* __restrict__ w0, const float* __restrict__ b0,
                      const float* __restrict__ Wd, const float* __restrict__ bk,
                      float* __restrict__ phi)
{
  int i = blockIdx.x*blockDim.x + threadIdx.x;
  if (i >= Bb*24) return;
  int b = i/24, t = i%24;
  float tau[DMm];
  for (int d = 0; d < DMm; ++d)
    tau[d] = (t < Ll) ? seqs[(b*DMm + d)*Ll + t] : tgts[(b*DMm + d)*Pp + (t - Ll)];
  float p = b0[0];
  for (int d = 0; d < DMm; ++d) p += tau[d]*w0[d];
  for (int k = 0; k < Kk16; ++k) {
    float s = bk[k];
    for (int d = 0; d < DMm; ++d) s += tau[d]*Wd[k*DMm + d];
    p += sinf(s);
  }
  phi[i] = p;
}

// collapse fus_W/glu into G1(9),g1(3),G2(9),g2(3)
__global__ void k_G(const float* __restrict__ fusW, const float* __restrict__ fusb,
                    const float* __restrict__ g1W, const float* __restrict__ g1b,
                    const float* __restrict__ g2W, const float* __restrict__ g2b,
                    float* __restrict__ G)
{
  int t = threadIdx.x;
  if (t >= 24) return;
  if (t < 9) {
    int c = t/3, o = t%3; float s = 0.f;
    for (int d = 0; d < Dd; ++d) s += fusW[c*Dd + d]*g1W[d*3 + o];
    G[t] = s;
  } else if (t < 12) {
    int o = t - 9; float s = g1b[o];
    for (int d = 0; d < Dd; ++d) s += fusb[d]*g1W[d*3 + o];
    G[9 + o] = s;
  } else if (t < 21) {
    int q = t - 12, c = q/3, o = q%3; float s = 0.f;
    for (int d = 0; d < Dd; ++d) s += fusW[c*Dd + d]*g2W[d*3 + o];
    G[12 + q] = s;
  } else {
    int o = t - 21; float s = g2b[o];
    for (int d = 0; d < Dd; ++d) s += fusb[d]*g2W[d*3 + o];
    G[21 + o] = s;
  }
}

// e1 = MB@We1, e2 = MB@We2 (one block per node)
__global__ __launch_bounds__(128)
void k_e(const float* __restrict__ mb, const float* __restrict__ We1,
         const float* __restrict__ We2, float* __restrict__ e1, float* __restrict__ e2)
{
  int n = blockIdx.x, d = threadIdx.x;
  __shared__ float row[Dd];
  row[d] = mb[n*Dd + d];
  __syncthreads();
  float s1 = 0.f, s2 = 0.f;
  for (int k = 0; k < Dd; ++k) { float m = row[k]; s1 += m*We1[k*Dd + d]; s2 += m*We2[k*Dd + d]; }
  e1[n*Dd + d] = s1; e2[n*Dd + d] = s2;
}

// a = e1 @ e2^T
__global__ __launch_bounds__(256)
void k_a(const float* __restrict__ e1, const float* __restrict__ e2, float* __restrict__ a)
{
  int i = blockIdx.x, tid = threadIdx.x;
  __shared__ float row[Dd];
  if (tid < Dd) row[tid] = e1[i*Dd + tid];
  __syncthreads();
  for (int j = tid; j < Nn; j += 256) {
    float s = 0.f;
    const float* p = e2 + j*Dd;
    for (int k = 0; k < Dd; ++k) s += row[k]*p[k];
    a[(size_t)i*Nn + j] = s;
  }
}

// per-row standardize (ddof=1) + relu + (-1e6 on diag) + softmax, in place
__global__ __launch_bounds__(256)
void k_rownorm(float* __restrict__ a)
{
  int i = blockIdx.x, tid = threadIdx.x;
  __shared__ float s[Nn];
  __shared__ float red[256];
  __shared__ float bc[2];
  for (int j = tid; j < Nn; j += 256) s[j] = a[(size_t)i*Nn + j];
  __syncthreads();
  float loc = 0.f;
  for (int j = tid; j < Nn; j += 256) loc += s[j];
  red[tid] = loc; __syncthreads();
  for (int o = 128; o > 0; o >>= 1) { if (tid < o) red[tid] += red[tid+o]; __syncthreads(); }
  if (tid == 0) bc[0] = red[0] / (float)Nn;
  __syncthreads();
  float mean = bc[0];
  loc = 0.f;
  for (int j = tid; j < Nn; j += 256) { float d = s[j]-mean; loc += d*d; }
  red[tid] = loc; __syncthreads();
  for (int o = 128; o > 0; o >>= 1) { if (tid < o) red[tid] += red[tid+o]; __syncthreads(); }
  if (tid == 0) bc[1] = sqrtf(red[0] / (float)(Nn-1));
  __syncthreads();
  float inv = 1.f / bc[1];
  for (int j = tid; j < Nn; j += 256) {
    float z = fmaxf((s[j]-mean)*inv, 0.f);
    if (j == i) z -= 1e6f;
    s[j] = z;
  }
  __syncthreads();
  loc = -3.4e38f;
  for (int j = tid; j < Nn; j += 256) loc = fmaxf(loc, s[j]);
  red[tid] = loc; __syncthreads();
  for (int o = 128; o > 0; o >>= 1) { if (tid < o) red[tid] = fmaxf(red[tid], red[tid+o]); __syncthreads(); }
  if (tid == 0) bc[0] = red[0];
  __syncthreads();
  float mx = bc[0];
  loc = 0.f;
  for (int j = tid; j < Nn; j += 256) { float e = expf(s[j]-mx); s[j] = e; loc += e; }
  red[tid] = loc; __syncthreads();
  for (int o = 128; o > 0; o >>= 1) { if (tid < o) red[tid] += red[tid+o]; __syncthreads(); }
  if (tid == 0) bc[1] = 1.f / red[0];
  __syncthreads();
  float dn = bc[1];
  for (int j = tid; j < Nn; j += 256) a[(size_t)i*Nn + j] = s[j]*dn;
}

// top-9 per row (values + indices), lax.top_k tie-break (lower index wins)
__global__ __launch_bounds__(128)
void k_topk(const float* __restrict__ a, float* __restrict__ topv, int* __restrict__ topi)
{
  int i = blockIdx.x, tid = threadIdx.x;
  __shared__ float s[Nn];
  __shared__ float rv[128];
  __shared__ int   ri[128];
  for (int j = tid; j < Nn; j += 128) s[j] = a[(size_t)i*Nn + j];
  __syncthreads();
  for (int it = 0; it < Mm-1; ++it) {
    float bv = -3.4e38f; int bi = Nn;
    for (int j = tid; j < Nn; j += 128) {
      float v = s[j];
      if (v > bv) { bv = v; bi = j; }
    }
    rv[tid] = bv; ri[tid] = bi; __syncthreads();
    for (int o = 64; o > 0; o >>= 1) {
      if (tid < o) {
        if (rv[tid+o] > rv[tid] || (rv[tid+o] == rv[tid] && ri[tid+o] < ri[tid])) {
          rv[tid] = rv[tid+o]; ri[tid] = ri[tid+o];
        }
      }
      __syncthreads();
    }
    if (tid == 0) { topv[i*(Mm-1)+it] = rv[0]; topi[i*(Mm-1)+it] = ri[0]; s[ri[0]] = -3.4e38f; }
    __syncthreads();
  }
}

// t = bn1(regR) + conv_in(xb) -> xbuf; partial stats of t per channel
__global__ __launch_bounds__(256)
void k_make_t(const float* __restrict__ reg, const float* __restrict__ xb,
              const float* __restrict__ st1, const float* __restrict__ cw,
              const float* __restrict__ cb, float* __restrict__ xbuf,
              float* __restrict__ partial)
{
  int c = blockIdx.x, tid = threadIdx.x;
  float s = 0.f, s2 = 0.f;
  int stride = gridDim.y * blockDim.x;
  for (int i = blockIdx.y*blockDim.x + tid; i < CNT; i += stride) {
    int b = i/NL, r = i%NL, n = r/Ll, l = r%Ll;
    float v = (reg[(size_t)(c*Ll + l)*NCP + b*Nn + n] - st1[c]) * st1[3+c];
    size_t base = (size_t)(b*Cc)*NL + r;
    v += cw[c*3+0]*xb[base] + cw[c*3+1]*xb[base+NL] + cw[c*3+2]*xb[base+2*NL] + cb[c];
    xbuf[(size_t)(b*Cc + c)*NL + r] = v;
    s += v; s2 += v*v;
  }
  __shared__ float sh[256], sh2[256];
  sh[tid] = s; sh2[tid] = s2; __syncthreads();
  for (int o = 128; o > 0; o >>= 1) {
    if (tid < o) { sh[tid] += sh[tid+o]; sh2[tid] += sh2[tid+o]; }
    __syncthreads();
  }
  if (tid == 0) {
    partial[(c*gridDim.y + blockIdx.y)*2 + 0] = sh[0];
    partial[(c*gridDim.y + blockIdx.y)*2 + 1] = sh2[0];
  }
}

__global__ void k_norm_x(float* __restrict__ x, const float* __restrict__ st)
{
  int i = blockIdx.x*blockDim.x + threadIdx.x;
  if (i >= Bb*Cc*NL) return;
  int c = (i % (Cc*NL)) / NL;
  x[i] = (x[i] - st[c]) * st[3+c];
}

// ===================== fused D2V attention head ========================
__global__ __launch_bounds__(128)
void k_head(const float* __restrict__ x, const float* __restrict__ phi,
            const float* __restrict__ topv, const int* __restrict__ topi,
            const float* __restrict__ wie, const float* __restrict__ mixp,
            const float* __restrict__ G, float* __restrict__ out)
{
  int bn = blockIdx.x;
  int b = bn / Nn, n = bn % Nn;
  int tid = threadIdx.x;
  __shared__ float xe[Cc][Mm][Ll];
  __shared__ float g[Cc][Mm];
  __shared__ float sc[Mm][Pp][Ll];
  __shared__ float pm[Cc][Mm][Pp];
  __shared__ float pr[Cc][Pp];
  __shared__ float wv[Mm];
  __shared__ float ph[24];
  __shared__ int   nbr[Mm];

  if (tid < 24) ph[tid] = phi[b*24 + tid];
  if (tid < Mm) {
    nbr[tid] = (tid == 0) ? n : topi[n*(Mm-1) + tid - 1];
    wv[tid]  = ((tid == 0) ? 1.f : topv[n*(Mm-1) + tid - 1]) * wie[tid];
  }
  __syncthreads();
  for (int i = tid; i < Cc*Mm*Ll; i += 128) {
    int c = i/(Mm*Ll), r = i%(Mm*Ll), m = r/Ll, l = r%Ll;
    xe[c][m][l] = x[((size_t)(b*Cc + c)*Nn + nbr[m])*Ll + l];
  }
  __syncthreads();
  float mix = mixp[0];
  if (tid < Cc*Mm) {
    int c = tid/Mm, m = tid%Mm;
    float s = 0.f;
    for (int l = 0; l < Ll; ++l) s += xe[c][m][l];
    g[c][m] = mix * (s / (float)Ll);
  }
  __syncthreads();
  for (int i = tid; i < Mm*Pp*Ll; i += 128) {
    int m = i/(Pp*Ll), r = i%(Pp*Ll), p = r/Ll, l = r%Ll;
    float s = 0.f;
    for (int c = 0; c < Cc; ++c) s += (ph[Ll + p] + g[c][m]) * (ph[l] + g[c][m]);
    sc[m][p][l] = s * (1.f/36.f);
  }
  __syncthreads();
  if (tid < Mm*Pp) {
    int m = tid/Pp, p = tid%Pp;
    float mx = -3.4e38f;
    for (int l = 0; l < Ll; ++l) mx = fmaxf(mx, sc[m][p][l]);
    float s = 0.f;
    for (int l = 0; l < Ll; ++l) { float e = expf(sc[m][p][l] - mx); sc[m][p][l] = e; s += e; }
    float inv = 1.f / s;
    for (int l = 0; l < Ll; ++l) sc[m][p][l] *= inv;
  }
  __syncthreads();
  for (int i = tid; i < Cc*Mm*Pp; i += 128) {
    int c = i/(Mm*Pp), r = i%(Mm*Pp), m = r/Pp, p = r%Pp;
    float s = 0.f;
    for (int l = 0; l < Ll; ++l) s += sc[m][p][l]*xe[c][m][l];
    pm[c][m][p] = s;
  }
  __syncthreads();
  if (tid < Cc*Pp) {
    int c = tid/Pp, p = tid%Pp;
    float s = 0.f;
    for (int m = 0; m < Mm; ++m) s += pm[c][m][p]*wv[m];
    pr[c][p] = s;
  }
  __syncthreads();
  if (tid < Cc*Pp) {
    int o = tid/Pp, p = tid%Pp;
    float o1 = G[9 + o], o2 = G[21 + o];
    for (int c = 0; c < Cc; ++c) { o1 += pr[c][p]*G[c*3 + o]; o2 += pr[c][p]*G[12 + c*3 + o]; }
    out[((size_t)(b*Cc + o)*Nn + n)*Pp + p] = o1 / (1.f + expf(-o2));
  }
}

// =====================================================================
extern "C" void kernel_launch(void* const* d_in, const int* in_sizes, int n_in,
                              void* d_out, int out_size, void* d_ws, size_t ws_size,
                              hipStream_t stream) {
  (void)in_sizes; (void)n_in; (void)out_size; (void)ws_size;
  const float* input = (const float*)d_in[0];
  const float* seqs  = (const float*)d_in[2];
  const float* tgts  = (const float*)d_in[3];
  const float* mb    = (const float*)d_in[4];
  const float* We1   = (const float*)d_in[5];
  const float* We2   = (const float*)d_in[6];
  const float* tde   = (const float*)d_in[7];
  const float* dwe   = (const float*)d_in[8];
  const float* tsW   = (const float*)d_in[9];
  const float* tsb   = (const float*)d_in[10];
  const float* cw    = (const float*)d_in[11];
  const float* cb    = (const float*)d_in[12];
  const float* encW1 = (const float*)d_in[13];
  const float* encb1 = (const float*)d_in[14];
  const float* encW2 = (const float*)d_in[15];
  const float* encb2 = (const float*)d_in[16];
  const float* regW  = (const float*)d_in[17];
  const float* regb  = (const float*)d_in[18];
  const float* d2vw0 = (const float*)d_in[19];
  const float* d2vb0 = (const float*)d_in[20];
  const float* d2vW  = (const float*)d_in[21];
  const float* d2vb  = (const float*)d_in[22];
  const float* d2vmx = (const float*)d_in[23];
  const float* wie   = (const float*)d_in[24];
  const float* fusW  = (const float*)d_in[25];
  const float* fusb  = (const float*)d_in[26];
  const float* gW1   = (const float*)d_in[27];
  const float* gb1   = (const float*)d_in[28];
  const float* gW2   = (const float*)d_in[29];
  const float* gb2   = (const float*)d_in[30];
  float* out = (float*)d_out;
  float* ws  = (float*)d_ws;

  // workspace layout (floats)
  const size_t SZ_ELT = (size_t)Bb*Cc*NL;        // 1,017,216
  const size_t SZ_HID = (size_t)Hh*NCP;          // 512 x 28288
  const size_t SZ_R64 = (size_t)64*NCP;          // 64 x 28288 (ts36 / reg shared)
  float* xb    = ws;                    // (b,c,n,l)
  float* ts36  = xb   + SZ_ELT;         // 36 x NCP (later reused as 64 x NCP reg)
  float* hidden= ts36 + SZ_R64;         // 512 x NCP
  float* hbuf  = hidden + SZ_HID;       // 512 x NCP
  float* xbuf  = hbuf + SZ_HID;         // (b,c,n,l) -> x
  float* e1    = xbuf + SZ_ELT;         // 883 x 128
  float* e2    = e1 + (size_t)Nn*Dd;
  float* amat  = e2 + (size_t)Nn*Dd;    // 883 x 883
  float* topv  = amat + (size_t)Nn*Nn;  // 883 x 9
  int*   topi  = (int*)(topv + (size_t)Nn*(Mm-1));
  float* phi   = (float*)(topi + (size_t)Nn*(Mm-1));  // 32 x 24
  int*   idx   = (int*)(phi + Bb*24);   // tidx[32], didx[32]
  float* part  = (float*)(idx + 64);    // 3*64*2 partial sums
  float* st_in = part + 3*64*2;         // mean[3],inv[3]
  float* st_rg = st_in + 6;
  float* st_t  = st_rg + 6;
  float* G     = st_t + 6;              // 24
  float* regWp = G + 24;                // 64 x 512 zero-padded reg weight
  float* regbp = regWp + 64*Hh;         // 64 padded reg bias
  float* regB  = ts36;                  // reg output reuses ts36 region

  const int NB = 64;                    // stats blocks per channel
  const int ntN = NCP / 64;             // 442 N-tiles for WMMA GEMMs

  // --- graph branch: e1/e2 -> a -> rownorm/softmax -> top-k ---
  k_tidx<<<1, 32, 0, stream>>>(seqs, idx);
  k_phi<<<(Bb*24 + 127)/128, 128, 0, stream>>>(seqs, tgts, d2vw0, d2vb0, d2vW, d2vb, phi);
  k_G<<<1, 32, 0, stream>>>(fusW, fusb, gW1, gb1, gW2, gb2, G);
  k_pad_reg<<<(64*Hh + 255)/256, 256, 0, stream>>>(regW, regb, regWp, regbp);
  k_e<<<Nn, Dd, 0, stream>>>(mb, We1, We2, e1, e2);
  k_a<<<Nn, 256, 0, stream>>>(e1, e2, amat);
  k_rownorm<<<Nn, 256, 0, stream>>>(amat);
  k_topk<<<Nn, 128, 0, stream>>>(amat, topv, topi);

  // --- BN(input) -> xb + ts36 ---
  k_chan_stats<<<dim3(Cc, NB), 256, 0, stream>>>(input, part, 0);
  k_stats_fin<<<Cc, 64, 0, stream>>>(part, st_in, NB);
  k_norm_in<<<(int)((SZ_ELT + 255)/256), 256, 0, stream>>>(input, st_in, xb, ts36);

  // --- hidden assembly ---
  k_wmma_gemm<<<dim3(ntN, Dd/32), 32, 0, stream>>>(tsW, ts36, hidden, tsb, Cc*Ll, 0);
  k_fill_hidden<<<(384*NC + 255)/256, 256, 0, stream>>>(hidden, mb, tde, dwe, idx);

  // --- residual encoder: 3 layers of relu-GEMM + residual-GEMM ---
  for (int i = 0; i < 3; ++i) {
    k_wmma_gemm<<<dim3(ntN, Hh/32), 32, 0, stream>>>(encW1 + (size_t)i*Hh*Hh, hidden, hbuf,
                                                     encb1 + i*Hh, Hh, 1);
    k_wmma_gemm<<<dim3(ntN, Hh/32), 32, 0, stream>>>(encW2 + (size_t)i*Hh*Hh, hbuf, hidden,
                                                     encb2 + i*Hh, Hh, 2);
  }

  // --- regression head (padded 64 x 512) ---
  k_wmma_gemm<<<dim3(ntN, 2), 32, 0, stream>>>(regWp, hidden, regB, regbp, Hh, 0);

  // --- BN(regR), conv mix, BN(t) -> x ---
  k_chan_stats<<<dim3(Cc, NB), 256, 0, stream>>>(regB, part, 1);
  k_stats_fin<<<Cc, 64, 0, stream>>>(part, st_rg, NB);
  k_make_t<<<dim3(Cc, NB), 256, 0, stream>>>(regB, xb, st_rg, cw, cb, xbuf, part);
  k_stats_fin<<<Cc, 64, 0, stream>>>(part, st_t, NB);
  k_norm_x<<<(int)((SZ_ELT + 255)/256), 256, 0, stream>>>(xbuf, st_t);

  // --- fused D2V attention + fus/GLU output ---
  k_head<<<Bb*Nn, 128, 0, stream>>>(xbuf, phi, topv, topi, wie, d2vmx, G, out);
}